// DynamicSimpleVFEWithVirtual_26834955665465
// MI455X (gfx1250) — compile-verified
//
#include <hip/hip_runtime.h>
#include <stdint.h>

#define FDIM 18
#define OUTC 21
#define KEYBITS 24
#define KEYSPACE (1u << KEYBITS)

#define SCAN_BLOCK 256
#define SCAN_ITEMS 8
#define SCAN_CHUNK (SCAN_BLOCK * SCAN_ITEMS)      // 2048
#define SCAN_NBLOCKS (KEYSPACE / SCAN_CHUNK)      // 8192

// gfx1250 async Global->LDS DMA path (ASYNCcnt), with safe fallback
#if defined(__AMDGCN__) && defined(__gfx1250__)
#if __has_builtin(__builtin_amdgcn_global_load_async_to_lds_b64) && \
    __has_builtin(__builtin_amdgcn_s_wait_asynccnt)
#define USE_ASYNC_LDS 1
#endif
#endif

#ifdef USE_ASYNC_LDS
typedef int v2i __attribute__((ext_vector_type(2)));
typedef __attribute__((address_space(1))) v2i as1_v2i;   // global
typedef __attribute__((address_space(3))) v2i as3_v2i;   // LDS
#endif

// ---------------- fills ----------------
__global__ void k_fill_u32(uint32_t* __restrict__ p, uint32_t v, uint32_t n) {
    uint32_t i = blockIdx.x * blockDim.x + threadIdx.x;
    uint32_t stride = gridDim.x * blockDim.x;
    for (; i < n; i += stride) p[i] = v;
}

// ---------------- mark presence ----------------
__global__ void k_mark(const int* __restrict__ coors, uint8_t* __restrict__ flags, int n) {
    int i = blockIdx.x * blockDim.x + threadIdx.x;
    if (i >= n) return;
    int4 c = ((const int4*)coors)[i];   // row stride 16B, aligned
    uint32_t key = ((uint32_t)c.x << 18) | ((uint32_t)c.y << 12) |
                   ((uint32_t)c.z << 6) | (uint32_t)c.w;
    flags[key] = 1;
}

// ---- helper: fetch this thread's 8 flag bytes (async-DMA to LDS when available) ----
__device__ __forceinline__ void load_flags8(const uint8_t* __restrict__ flags,
                                            uint32_t base, uint8_t* lbuf,
                                            uint32_t& a, uint32_t& b) {
#ifdef USE_ASYNC_LDS
    __builtin_amdgcn_global_load_async_to_lds_b64(
        (as1_v2i*)(flags + base),
        (as3_v2i*)(lbuf + (size_t)threadIdx.x * SCAN_ITEMS),
        0, 0);
    __builtin_amdgcn_s_wait_asynccnt(0);
    const uint32_t* lp = (const uint32_t*)(lbuf + (size_t)threadIdx.x * SCAN_ITEMS);
    a = lp[0];
    b = lp[1];
#else
    (void)lbuf;
    const uint32_t* p = (const uint32_t*)(flags + base);   // 8B aligned
    a = p[0];
    b = p[1];
#endif
}

// ---------------- scan pass 1: per-block sums ----------------
__global__ void k_scan_blocksums(const uint8_t* __restrict__ flags,
                                 uint32_t* __restrict__ blockSums) {
    __shared__ uint32_t lds[SCAN_BLOCK];
    __shared__ uint8_t fbuf[SCAN_CHUNK];
    uint32_t base = blockIdx.x * SCAN_CHUNK + threadIdx.x * SCAN_ITEMS;
    uint32_t a, b;
    load_flags8(flags, base, fbuf, a, b);
    // flag bytes are 0/1 -> popcount == byte sum
    uint32_t s = __popc(a) + __popc(b);
    lds[threadIdx.x] = s;
    __syncthreads();
    for (int off = SCAN_BLOCK / 2; off > 0; off >>= 1) {
        if ((int)threadIdx.x < off) lds[threadIdx.x] += lds[threadIdx.x + off];
        __syncthreads();
    }
    if (threadIdx.x == 0) blockSums[blockIdx.x] = lds[0];
}

// ---------------- scan pass 2: single-block scan of 8192 block sums ----------------
__global__ void k_scan_offsets(uint32_t* __restrict__ blockSums) {
    __shared__ uint32_t lds[256];
    const int PER = SCAN_NBLOCKS / 256;   // 32
    int t = threadIdx.x;
    uint32_t local[PER];
    uint32_t sum = 0;
    int base = t * PER;
    #pragma unroll
    for (int i = 0; i < PER; ++i) { local[i] = blockSums[base + i]; sum += local[i]; }
    lds[t] = sum;
    __syncthreads();
    // Hillis-Steele inclusive scan over 256
    for (int off = 1; off < 256; off <<= 1) {
        uint32_t v = (t >= off) ? lds[t - off] : 0u;
        __syncthreads();
        lds[t] += v;
        __syncthreads();
    }
    uint32_t run = (t == 0) ? 0u : lds[t - 1];   // exclusive
    #pragma unroll
    for (int i = 0; i < PER; ++i) { uint32_t v = local[i]; blockSums[base + i] = run; run += v; }
}

// ---------------- scan pass 3: write exclusive ranks ----------------
__global__ void k_scan_ranks(const uint8_t* __restrict__ flags,
                             const uint32_t* __restrict__ blockSums,
                             uint32_t* __restrict__ rank) {
    __shared__ uint32_t lds[SCAN_BLOCK];
    __shared__ uint8_t fbuf[SCAN_CHUNK];
    int t = threadIdx.x;
    uint32_t base = blockIdx.x * SCAN_CHUNK + t * SCAN_ITEMS;
    uint32_t a, b;
    load_flags8(flags, base, fbuf, a, b);
    uint32_t s = __popc(a) + __popc(b);
    lds[t] = s;
    __syncthreads();
    for (int off = 1; off < SCAN_BLOCK; off <<= 1) {
        uint32_t v = (t >= off) ? lds[t - off] : 0u;
        __syncthreads();
        lds[t] += v;
        __syncthreads();
    }
    uint32_t run = ((t == 0) ? 0u : lds[t - 1]) + blockSums[blockIdx.x];
    uint32_t r[SCAN_ITEMS];
    #pragma unroll
    for (int j = 0; j < 4; ++j) { r[j] = run; run += (a >> (8 * j)) & 1u; }
    #pragma unroll
    for (int j = 0; j < 4; ++j) { r[4 + j] = run; run += (b >> (8 * j)) & 1u; }
    uint4* dst = (uint4*)(rank + base);   // base multiple of 8 -> 32B aligned
    dst[0] = make_uint4(r[0], r[1], r[2], r[3]);
    dst[1] = make_uint4(r[4], r[5], r[6], r[7]);
}

// ---------------- scatter-accumulate ----------------
__global__ void k_scatter(const float* __restrict__ feat, const int* __restrict__ coors,
                          const uint32_t* __restrict__ rank,
                          float* __restrict__ out, float* __restrict__ counts, int n) {
    uint32_t tid = blockIdx.x * blockDim.x + threadIdx.x;
    uint32_t stride = gridDim.x * blockDim.x;
    for (uint32_t i = tid; i < (uint32_t)n; i += stride) {
        // cover next iteration's row latency (speculative: OOB prefetch is dropped)
        __builtin_prefetch(feat + (size_t)(i + stride) * FDIM, 0, 1);
        __builtin_prefetch(coors + (size_t)(i + stride) * 4, 0, 1);

        const float2* fp = (const float2*)(feat + (size_t)i * FDIM);  // 72B rows, 8B aligned
        float f[FDIM];
        #pragma unroll
        for (int j = 0; j < 9; ++j) { float2 v = fp[j]; f[2 * j] = v.x; f[2 * j + 1] = v.y; }

        int4 c = ((const int4*)coors)[i];
        uint32_t key = ((uint32_t)c.x << 18) | ((uint32_t)c.y << 12) |
                       ((uint32_t)c.z << 6) | (uint32_t)c.w;
        uint32_t r = rank[key];
        float* orow = out + (size_t)r * OUTC;

        float flag = f[16];
        bool real   = (flag == 1.0f);
        bool paintd = (flag == 0.0f);
        bool vmask  = paintd || (flag == -1.0f);

        if (real) {
            // REAL_DIMS = {0,1,2,3,17} -> out cols 0..4
            atomicAdd(orow + 0, f[0]);
            atomicAdd(orow + 1, f[1]);
            atomicAdd(orow + 2, f[2]);
            atomicAdd(orow + 3, f[3]);
            atomicAdd(orow + 4, f[17]);
            atomicAdd(counts + r, 1.0f);               // real count
        }
        if (vmask) {
            // VIRT_DIMS = {0..13,17} -> out cols 5..19 ; col 20 = painted count
            #pragma unroll
            for (int j = 0; j < 14; ++j) atomicAdd(orow + 5 + j, f[j]);
            atomicAdd(orow + 19, f[17]);
            if (paintd) atomicAdd(orow + 20, 1.0f);
            atomicAdd(counts + (size_t)n + r, 1.0f);   // virtual count
        }
    }
}

// ---------------- write sorted unique coords (4 keys / thread) ----------------
__global__ void k_write_uniq(const uint32_t* __restrict__ flagsW,
                             const uint32_t* __restrict__ rank,
                             float* __restrict__ uniq, uint32_t nwords) {
    uint32_t w = blockIdx.x * blockDim.x + threadIdx.x;
    if (w >= nwords) return;
    uint32_t word = flagsW[w];
    if (word == 0u) return;                       // ~59% of words: no key present
    uint4 rk = ((const uint4*)rank)[w];           // 16B-aligned rank quad
    uint32_t rr[4] = { rk.x, rk.y, rk.z, rk.w };
    uint32_t k0 = w * 4u;
    #pragma unroll
    for (int j = 0; j < 4; ++j) {
        if ((word >> (8 * j)) & 1u) {
            uint32_t k = k0 + (uint32_t)j;
            float* row = uniq + (size_t)rr[j] * 4;
            row[0] = (float)(k >> 18);
            row[1] = (float)((k >> 12) & 63u);
            row[2] = (float)((k >> 6) & 63u);
            row[3] = (float)(k & 63u);
        }
    }
}

// ---------------- normalize means (element-parallel, coalesced) ----------------
__global__ void k_norm(float* __restrict__ out, const float* __restrict__ counts, int n) {
    uint32_t e = blockIdx.x * blockDim.x + threadIdx.x;
    uint32_t total = (uint32_t)n * OUTC;
    if (e >= total) return;
    uint32_t r = e / OUTC;              // strength-reduced to mul-shift
    uint32_t c = e - r * OUTC;
    float cnt = (c < 5) ? counts[r] : counts[(size_t)n + r];
    out[e] = out[e] / fmaxf(cnt, 1.0f);   // true divide to match reference
}

extern "C" void kernel_launch(void* const* d_in, const int* in_sizes, int n_in,
                              void* d_out, int out_size, void* d_ws, size_t ws_size,
                              hipStream_t stream) {
    const float* feat  = (const float*)d_in[0];
    const int*   coors = (const int*)d_in[1];
    int n = in_sizes[0] / FDIM;                // 2,000,000

    float* out  = (float*)d_out;               // n * 21
    float* uniq = out + (size_t)n * OUTC;      // n * 4 (coords emitted as float values, pad = -1)

    uint8_t*  flags     = (uint8_t*)d_ws;                                   // 16 MB
    uint32_t* rank      = (uint32_t*)(flags + (size_t)KEYSPACE);            // 64 MB
    float*    counts    = (float*)((char*)rank + (size_t)KEYSPACE * 4);     // 2n floats
    uint32_t* blockSums = (uint32_t*)((char*)counts + (size_t)2 * n * 4);   // 32 KB

    const int FB = 256, FG = 4096;
    k_fill_u32<<<FG, FB, 0, stream>>>((uint32_t*)flags, 0u, KEYSPACE / 4);
    k_fill_u32<<<FG, FB, 0, stream>>>((uint32_t*)counts, 0u, (uint32_t)(2 * n));
    k_fill_u32<<<FG, FB, 0, stream>>>((uint32_t*)out, 0u, (uint32_t)((size_t)n * OUTC));
    k_fill_u32<<<FG, FB, 0, stream>>>((uint32_t*)uniq, 0xBF800000u /* -1.0f */, (uint32_t)(4 * n));

    k_mark<<<(n + 255) / 256, 256, 0, stream>>>(coors, flags, n);

    k_scan_blocksums<<<SCAN_NBLOCKS, SCAN_BLOCK, 0, stream>>>(flags, blockSums);
    k_scan_offsets<<<1, 256, 0, stream>>>(blockSums);
    k_scan_ranks<<<SCAN_NBLOCKS, SCAN_BLOCK, 0, stream>>>(flags, blockSums, rank);

    k_scatter<<<2048, 256, 0, stream>>>(feat, coors, rank, out, counts, n);

    k_write_uniq<<<(KEYSPACE / 4 + 255) / 256, 256, 0, stream>>>(
        (const uint32_t*)flags, rank, uniq, KEYSPACE / 4);

    uint32_t totalElems = (uint32_t)((size_t)n * OUTC);
    k_norm<<<(totalElems + 255) / 256, 256, 0, stream>>>(out, counts, n);
}